// MMPD_DTA_57466662421180
// MI455X (gfx1250) — compile-verified
//
#include <hip/hip_runtime.h>
#include <hip/hip_bf16.h>
#include <math.h>
#include <stdint.h>

// ---------------------------------------------------------------------------
// CDNA5 WMMA / TDM types
// ---------------------------------------------------------------------------
typedef float v2f __attribute__((ext_vector_type(2)));
typedef float v8f __attribute__((ext_vector_type(8)));
typedef unsigned int u32x4 __attribute__((ext_vector_type(4)));
typedef int i32x4 __attribute__((ext_vector_type(4)));
typedef int i32x8 __attribute__((ext_vector_type(8)));

#define GEMM_BN 128   // N columns per block
#define GEMM_KT 64    // K tile staged in LDS

// ---------------------------------------------------------------------------
// Generic fp32 GEMM with V_WMMA_F32_16X16X4_F32 + TDM-staged W tiles.
// C[M,N] = act( A[M,K] @ W[K,N] + bias (+ C if accum) )
// act: 0=none, 1=relu, 2=prelu(*alphap)
// Block: 256 threads = 8 waves; wave w owns M-tile (bx*8+w)*16.
// W tile (64x128 f32, zero-padded OOB by TDM) DMA'd to LDS by wave 0 via
// tensor_load_to_lds; completion tracked with TENSORcnt.
// ---------------------------------------------------------------------------
__global__ __launch_bounds__(256)
void gemm_wmma_f32(const float* __restrict__ A, int lda,
                   const float* __restrict__ W, const float* __restrict__ bias,
                   float* __restrict__ C, int ldc,
                   int M, int N, int K,
                   int act, const float* __restrict__ alphap, int accum)
{
    __shared__ float Wl[GEMM_KT * GEMM_BN];   // 32 KB

    const int tid  = threadIdx.x;
    const int wave = tid >> 5;
    const int lane = tid & 31;
    const int m0   = (blockIdx.x * 8 + wave) * 16;
    const int n0   = blockIdx.y * GEMM_BN;

    const int mrow  = m0 + (lane & 15);   // A-matrix: lanes 0-15 M=0..15
    const int klsel = (lane >> 4) * 2;    // lanes 16-31 carry K+2,K+3
    const int ncol  = lane & 15;          // B/C/D: lanes 0-15 N=0..15

    // LDS byte offset of Wl: low 32 bits of the generic (flat) pointer.
    const unsigned int lds_base = (unsigned int)(uintptr_t)&Wl[0];
    // Wave-uniform wave id so the compiler can use a scalar branch (TDM
    // ignores EXEC; only one wave may issue the DMA).
    const int wave_u = __builtin_amdgcn_readfirstlane(wave);

    v8f acc[8];
    #pragma unroll
    for (int i = 0; i < 8; ++i)
        #pragma unroll
        for (int j = 0; j < 8; ++j) acc[i][j] = 0.0f;

    for (int kc = 0; kc < K; kc += GEMM_KT) {
        __syncthreads();                       // previous tile fully consumed
        if (wave_u == 0) {
            // ---- Tensor DMA descriptor (D#) ----
            const unsigned long long ga =
                (unsigned long long)(uintptr_t)(W + (size_t)kc * N + n0);
            u32x4 g0;
            g0[0] = 1u;                                   // count=1 (valid), no gather
            g0[1] = lds_base;                             // lds_addr (bytes)
            g0[2] = (unsigned int)ga;                     // global_addr[31:0]
            g0[3] = (unsigned int)((ga >> 32) & 0x01FFFFFFu) | (2u << 30); // addr[56:32] | type=2

            const unsigned int td0 = (unsigned int)(N - n0);  // remaining cols
            const unsigned int td1 = (unsigned int)(K - kc);  // remaining rows
            i32x8 g1;
            g1[0] = (int)(2u << 16);                      // wg_mask=0, data_size=2 (4B)
            g1[1] = (int)((td0 & 0xFFFFu) << 16);         // tensor_dim0[15:0] @ bits 63:48
            g1[2] = (int)((td0 >> 16) | ((td1 & 0xFFFFu) << 16)); // dim0 hi | dim1 lo
            g1[3] = (int)((td1 >> 16) | (128u << 16));    // dim1 hi | tile_dim0=128
            g1[4] = 64;                                   // tile_dim1=64, tile_dim2=0
            g1[5] = N;                                    // dim0_stride[31:0] (elements)
            g1[6] = 0;                                    // dim0_stride hi | dim1_stride lo
            g1[7] = 0;
            const i32x4 gz4 = {0, 0, 0, 0};               // groups 2/3 unused (2-D)
            const i32x8 gz8 = {0, 0, 0, 0, 0, 0, 0, 0};   // extra group (unused)
            __builtin_amdgcn_tensor_load_to_lds(g0, g1, gz4, gz4, gz8, 0);
            __builtin_amdgcn_s_wait_tensorcnt(0);
        }
        __syncthreads();                       // tile visible to all waves

        const int ksteps = (K - kc < GEMM_KT) ? (K - kc) : GEMM_KT;
        for (int k4 = 0; k4 * 4 < ksteps; ++k4) {
            const int gk = kc + k4 * 4 + klsel;
            v2f a;
            a.x = (mrow < M && gk     < K) ? A[(size_t)mrow * lda + gk]     : 0.0f;
            a.y = (mrow < M && gk + 1 < K) ? A[(size_t)mrow * lda + gk + 1] : 0.0f;
            const int lk = k4 * 4 + klsel;
            #pragma unroll
            for (int nt = 0; nt < 8; ++nt) {
                v2f b;
                b.x = Wl[lk       * GEMM_BN + nt * 16 + ncol];
                b.y = Wl[(lk + 1) * GEMM_BN + nt * 16 + ncol];
                acc[nt] = __builtin_amdgcn_wmma_f32_16x16x4_f32(
                    false, a, false, b, (short)0, acc[nt], false, false);
            }
        }
    }

    const float alpha = (act == 2 && alphap) ? *alphap : 0.0f;
    const int mbase = m0 + ((lane >> 4) ? 8 : 0);  // C/D: VGPR j -> M=j (+8 hi half)
    #pragma unroll
    for (int nt = 0; nt < 8; ++nt) {
        const int n = n0 + nt * 16 + ncol;
        if (n >= N) continue;
        const float bv = bias ? bias[n] : 0.0f;
        #pragma unroll
        for (int j = 0; j < 8; ++j) {
            const int m = mbase + j;
            if (m >= M) continue;
            float v = acc[nt][j] + bv;
            if (accum) v += C[(size_t)m * ldc + n];
            if (act == 1)      v = fmaxf(v, 0.0f);
            else if (act == 2) v = (v >= 0.0f) ? v : alpha * v;
            C[(size_t)m * ldc + n] = v;
        }
    }
}

// ---------------------------------------------------------------------------
// Elementwise / graph helper kernels
// ---------------------------------------------------------------------------
__global__ void kzero(float* p, long long n) {
    for (long long i = (long long)blockIdx.x * blockDim.x + threadIdx.x;
         i < n; i += (long long)gridDim.x * blockDim.x) p[i] = 0.0f;
}

// agg[dst[e], f] += x[src[e], f]   (row stride == F, packed)
__global__ void kscatter(const float* __restrict__ x, const int* __restrict__ src,
                         const int* __restrict__ dst, int E, int F,
                         float* __restrict__ agg) {
    const long long n = (long long)E * F;
    for (long long i = (long long)blockIdx.x * blockDim.x + threadIdx.x;
         i < n; i += (long long)gridDim.x * blockDim.x) {
        const int e = (int)(i / F);
        const int f = (int)(i % F);
        atomicAdd(&agg[(size_t)dst[e] * F + f], x[(size_t)src[e] * F + f]);
    }
}

__global__ void kadd_inplace(float* __restrict__ q, const float* __restrict__ x, long long n) {
    for (long long i = (long long)blockIdx.x * blockDim.x + threadIdx.x;
         i < n; i += (long long)gridDim.x * blockDim.x) q[i] += x[i];
}

__global__ void kcount(const int* __restrict__ dst, int E, float* __restrict__ cnt) {
    for (long long i = (long long)blockIdx.x * blockDim.x + threadIdx.x;
         i < E; i += (long long)gridDim.x * blockDim.x)
        atomicAdd(&cnt[dst[i]], 1.0f);
}

__global__ void krowdiv(float* __restrict__ q, const float* __restrict__ cnt,
                        long long M, int F) {
    const long long n = M * F;
    for (long long i = (long long)blockIdx.x * blockDim.x + threadIdx.x;
         i < n; i += (long long)gridDim.x * blockDim.x)
        q[i] /= fmaxf(cnt[i / F], 1.0f);
}

__global__ void kbnstats(const float* __restrict__ x, long long M, int F,
                         float* __restrict__ stats) {
    __shared__ float s1[128], s2[128];
    for (int i = threadIdx.x; i < F; i += blockDim.x) { s1[i] = 0.0f; s2[i] = 0.0f; }
    __syncthreads();
    const long long n = M * F;
    for (long long i = (long long)blockIdx.x * blockDim.x + threadIdx.x;
         i < n; i += (long long)gridDim.x * blockDim.x) {
        const float v = x[i];
        const int f = (int)(i % F);
        atomicAdd(&s1[f], v);
        atomicAdd(&s2[f], v * v);
    }
    __syncthreads();
    for (int i = threadIdx.x; i < F; i += blockDim.x) {
        atomicAdd(&stats[i], s1[i]);
        atomicAdd(&stats[F + i], s2[i]);
    }
}

__global__ void kbnapply(float* __restrict__ x, long long M, int F,
                         const float* __restrict__ stats,
                         const float* __restrict__ g, const float* __restrict__ b) {
    const long long n = M * F;
    const float invM = 1.0f / (float)M;
    for (long long i = (long long)blockIdx.x * blockDim.x + threadIdx.x;
         i < n; i += (long long)gridDim.x * blockDim.x) {
        const int f = (int)(i % F);
        const float mu = stats[f] * invM;
        const float var = stats[F + f] * invM - mu * mu;
        x[i] = g[f] * (x[i] - mu) * rsqrtf(var + 1e-5f) + b[f];
    }
}

__global__ void kpool(const float* __restrict__ x, const int* __restrict__ batch,
                      long long M, int F, float* __restrict__ out) {
    const long long n = M * F;
    for (long long i = (long long)blockIdx.x * blockDim.x + threadIdx.x;
         i < n; i += (long long)gridDim.x * blockDim.x) {
        const int r = (int)(i / F);
        const int f = (int)(i % F);
        atomicAdd(&out[(size_t)batch[r] * F + f], x[i]);
    }
}

// p[b,s,d] = src_emb[tok,d] + sinusoid(tok, d)   (ref indexes POS_TAB by token)
__global__ void kembed(const int* __restrict__ protein, const float* __restrict__ emb,
                       float* __restrict__ p, int total) {
    const int idx = blockIdx.x * blockDim.x + threadIdx.x;
    if (idx >= total) return;
    const int d  = idx % 60;
    const int bs = idx / 60;
    const int tok = protein[bs];
    const double e  = (double)(2 * (d / 2)) / 60.0;
    const double ang = (double)tok * pow(10000.0, -e);
    const float pe = (d & 1) ? (float)cos(ang) : (float)sin(ang);
    p[idx] = emb[tok * 60 + d] + pe;
}

// ---------------------------------------------------------------------------
// Transformer encoder: one block per sequence (S=26, D=60, H=2, dh=30)
// ---------------------------------------------------------------------------
struct TrP {
    const float *Wqkv, *bqkv, *Wo, *bo, *W1, *b1, *W2, *b2, *g1, *bb1, *g2, *bb2;
};

__global__ __launch_bounds__(64)
void tr_encoder(const float* __restrict__ pin, TrP p0, TrP p1,
                float* __restrict__ z, int ldz)
{
    __shared__ float X[26 * 60];
    __shared__ float QKV[26 * 180];
    __shared__ float OB[26 * 60];
    __shared__ float SC[26 * 26];
    __shared__ float F1[26 * 256];

    const int b = blockIdx.x, t = threadIdx.x;
    const float* src = pin + (size_t)b * 26 * 60;
    for (int i = t; i < 26 * 60; i += 64) X[i] = src[i];
    __syncthreads();

    TrP ps[2] = {p0, p1};
    for (int l = 0; l < 2; ++l) {
        TrP pp = ps[l];
        // qkv = x @ Wqkv + bqkv
        for (int i = t; i < 26 * 180; i += 64) {
            const int s = i / 180, j = i % 180;
            float acc = pp.bqkv[j];
            for (int k = 0; k < 60; ++k) acc += X[s * 60 + k] * pp.Wqkv[k * 180 + j];
            QKV[i] = acc;
        }
        __syncthreads();
        // attention per head
        for (int h = 0; h < 2; ++h) {
            for (int i = t; i < 26 * 26; i += 64) {
                const int qi = i / 26, ki = i % 26;
                float acc = 0.0f;
                for (int d = 0; d < 30; ++d)
                    acc += QKV[qi * 180 + h * 30 + d] * QKV[ki * 180 + 60 + h * 30 + d];
                SC[i] = acc * rsqrtf(30.0f);
            }
            __syncthreads();
            if (t < 26) {
                float mx = -1e30f;
                for (int k = 0; k < 26; ++k) mx = fmaxf(mx, SC[t * 26 + k]);
                float sm = 0.0f;
                for (int k = 0; k < 26; ++k) { float e = __expf(SC[t * 26 + k] - mx); SC[t * 26 + k] = e; sm += e; }
                const float inv = 1.0f / sm;
                for (int k = 0; k < 26; ++k) SC[t * 26 + k] *= inv;
            }
            __syncthreads();
            for (int i = t; i < 26 * 30; i += 64) {
                const int qi = i / 30, d = i % 30;
                float acc = 0.0f;
                for (int k = 0; k < 26; ++k)
                    acc += SC[qi * 26 + k] * QKV[k * 180 + 120 + h * 30 + d];
                OB[qi * 60 + h * 30 + d] = acc;
            }
            __syncthreads();
        }
        // o @ Wo + bo + residual -> F1 (scratch), then LN -> X
        for (int i = t; i < 26 * 60; i += 64) {
            const int s = i / 60, j = i % 60;
            float acc = pp.bo[j];
            for (int k = 0; k < 60; ++k) acc += OB[s * 60 + k] * pp.Wo[k * 60 + j];
            F1[i] = acc + X[i];
        }
        __syncthreads();
        if (t < 26) {
            float mu = 0.0f;
            for (int j = 0; j < 60; ++j) mu += F1[t * 60 + j];
            mu /= 60.0f;
            float v = 0.0f;
            for (int j = 0; j < 60; ++j) { float d = F1[t * 60 + j] - mu; v += d * d; }
            v /= 60.0f;
            const float inv = rsqrtf(v + 1e-5f);
            for (int j = 0; j < 60; ++j)
                X[t * 60 + j] = pp.g1[j] * (F1[t * 60 + j] - mu) * inv + pp.bb1[j];
        }
        __syncthreads();
        // FFN
        for (int i = t; i < 26 * 256; i += 64) {
            const int s = i / 256, j = i % 256;
            float acc = pp.b1[j];
            for (int k = 0; k < 60; ++k) acc += X[s * 60 + k] * pp.W1[k * 256 + j];
            F1[i] = fmaxf(acc, 0.0f);
        }
        __syncthreads();
        for (int i = t; i < 26 * 60; i += 64) {
            const int s = i / 60, j = i % 60;
            float acc = pp.b2[j];
            for (int k = 0; k < 256; ++k) acc += F1[s * 256 + k] * pp.W2[k * 60 + j];
            OB[i] = acc + X[i];
        }
        __syncthreads();
        if (t < 26) {
            float mu = 0.0f;
            for (int j = 0; j < 60; ++j) mu += OB[t * 60 + j];
            mu /= 60.0f;
            float v = 0.0f;
            for (int j = 0; j < 60; ++j) { float d = OB[t * 60 + j] - mu; v += d * d; }
            v /= 60.0f;
            const float inv = rsqrtf(v + 1e-5f);
            for (int j = 0; j < 60; ++j)
                X[t * 60 + j] = pp.g2[j] * (OB[t * 60 + j] - mu) * inv + pp.bb2[j];
        }
        __syncthreads();
    }
    for (int j = t; j < 60; j += 64) z[(size_t)b * ldz + j] = X[j];
}

// ---------------------------------------------------------------------------
// Host orchestration
// ---------------------------------------------------------------------------
extern "C" void kernel_launch(void* const* d_in, const int* in_sizes, int n_in,
                              void* d_out, int out_size, void* d_ws, size_t ws_size,
                              hipStream_t stream)
{
    const float* x_t  = (const float*)d_in[0];
    const int*   eT   = (const int*)  d_in[1];
    const int*   batT = (const int*)  d_in[2];
    const float* x_s  = (const float*)d_in[3];
    const int*   eS   = (const int*)  d_in[4];
    const int*   batS = (const int*)  d_in[5];
    const int*   prot = (const int*)  d_in[6];
    // Params flattened in setup_inputs() insertion order:
    // gin(16) bn(8) fc1(2) sage(9) sage_lin(2) tr(24) src_emb(1) cls(8)
    const float* const* pr = (const float* const*)(d_in + 7);

    const int Mn = in_sizes[0] / 108;   // 150000 nodes
    const int E  = in_sizes[1] / 2;     // 1200000 edges
    const int B  = 512;

    // Workspace layout (floats)
    float* ws = (float*)d_ws;
    const size_t NB = (size_t)Mn * 108;
    float* P = ws;
    float* Q = P + NB;
    float* R = Q + NB;
    float* sm = R + NB;
    float* pooledT = sm; sm += (size_t)B * 128;
    float* pooledS = sm; sm += (size_t)B * 128;
    float* z       = sm; sm += (size_t)B * 300;
    float* h1      = sm; sm += (size_t)B * 512;
    float* h2      = sm; sm += (size_t)B * 256;
    float* pemb    = sm; sm += (size_t)B * 26 * 60;
    float* cnt     = sm; sm += (size_t)Mn;
    float* stats   = sm; sm += 256;

    const int GS = 4096; // grid-stride blocks for big elementwise kernels
    auto zero = [&](float* p, long long n) {
        kzero<<<GS, 256, 0, stream>>>(p, n);
    };
    auto gemm = [&](const float* A, int lda, const float* W, const float* bias,
                    float* C, int ldc, int M, int N, int K,
                    int act, const float* alphap, int accum) {
        dim3 g((M + 127) / 128, (N + 127) / 128);
        gemm_wmma_f32<<<g, 256, 0, stream>>>(A, lda, W, bias, C, ldc, M, N, K,
                                             act, alphap, accum);
    };

    // ------------------- GIN branch -------------------
    const float* xin = x_t;
    for (int l = 0; l < 4; ++l) {
        const int F = (l == 0) ? 108 : 96;
        zero(Q, (long long)Mn * F);
        kscatter<<<GS, 256, 0, stream>>>(xin, eT, eT + E, E, F, Q);
        kadd_inplace<<<GS, 256, 0, stream>>>(Q, xin, (long long)Mn * F);     // h = x + agg
        gemm(Q, F,  pr[0 + 4 * l], pr[1 + 4 * l], R, 96, Mn, 96, F,  1, nullptr, 0); // relu(h@W1+b1)
        gemm(R, 96, pr[2 + 4 * l], pr[3 + 4 * l], P, 96, Mn, 96, 96, 1, nullptr, 0); // relu(.@W2+b2)
        zero(stats, 256);
        kbnstats<<<512, 256, 0, stream>>>(P, Mn, 96, stats);
        kbnapply<<<GS, 256, 0, stream>>>(P, Mn, 96, stats, pr[16 + 2 * l], pr[17 + 2 * l]);
        xin = P;
    }
    zero(pooledT, (long long)B * 96);
    kpool<<<GS, 256, 0, stream>>>(P, batT, Mn, 96, pooledT);
    gemm(pooledT, 96, pr[24], pr[25], z + 180, 300, B, 120, 96, 1, nullptr, 0);

    // ------------------- SAGE branch -------------------
    zero(cnt, Mn);
    kcount<<<GS, 256, 0, stream>>>(eS + E, E, cnt);
    const float* cin = x_s;
    float* out = P;
    for (int l = 0; l < 3; ++l) {
        const int F = (l == 0) ? 108 : 98;
        zero(Q, (long long)Mn * F);
        kscatter<<<GS, 256, 0, stream>>>(cin, eS, eS + E, E, F, Q);
        krowdiv<<<GS, 256, 0, stream>>>(Q, cnt, Mn, F);                       // agg / cnt
        out = (cin == P) ? R : P;
        gemm(Q,   F, pr[26 + 3 * l], pr[27 + 3 * l], out, 98, Mn, 98, F, 0, nullptr, 0); // agg@Wl+bl
        gemm(cin, F, pr[28 + 3 * l], nullptr,        out, 98, Mn, 98, F, 1, nullptr, 1); // += c@Wr, relu
        cin = out;
    }
    zero(pooledS, (long long)B * 98);
    kpool<<<GS, 256, 0, stream>>>(cin, batS, Mn, 98, pooledS);
    gemm(pooledS, 98, pr[35], pr[36], z + 60, 300, B, 120, 98, 0, nullptr, 0);

    // ------------------- Transformer branch -------------------
    const int etot = B * 26 * 60;
    kembed<<<(etot + 255) / 256, 256, 0, stream>>>(prot, pr[61], pemb, etot);
    TrP t0 = {pr[37], pr[38], pr[39], pr[40], pr[41], pr[42],
              pr[43], pr[44], pr[45], pr[46], pr[47], pr[48]};
    TrP t1 = {pr[49], pr[50], pr[51], pr[52], pr[53], pr[54],
              pr[55], pr[56], pr[57], pr[58], pr[59], pr[60]};
    tr_encoder<<<B, 64, 0, stream>>>(pemb, t0, t1, z, 300);

    // ------------------- Classifier -------------------
    gemm(z,  300, pr[62], pr[63], h1, 512, B, 512, 300, 2, pr[64], 0);
    gemm(h1, 512, pr[65], pr[66], h2, 256, B, 256, 512, 2, pr[67], 0);
    gemm(h2, 256, pr[68], pr[69], (float*)d_out, 1, B, 1, 256, 0, nullptr, 0);

    (void)n_in; (void)out_size; (void)ws_size;
}